// HybridClassifier_5385888989337
// MI455X (gfx1250) — compile-verified
//
#include <hip/hip_runtime.h>
#include <math.h>

typedef float v2f __attribute__((ext_vector_type(2)));
typedef float v8f __attribute__((ext_vector_type(8)));

#define NQ     4
#define DIM    16
#define KDIM   784
#define KB     (KDIM / 4)   // 196 K-blocks of 4
#define BLOCK  256

__global__ __launch_bounds__(BLOCK)
void hybrid_qnn_kernel(const float* __restrict__ x,
                       const float* __restrict__ W_red,
                       const float* __restrict__ b_red,
                       const float* __restrict__ qnn_w,
                       const float* __restrict__ W_out,
                       const float* __restrict__ b_out,
                       float* __restrict__ out)
{
    // B matrix (784x16, cols 4..15 zero) staged in WMMA B-fragment layout:
    // entry [kb*32 + lane]: lane<16 -> {B[4kb+0][lane],   B[4kb+1][lane]}
    //                       lane>=16-> {B[4kb+2][lane-16],B[4kb+3][lane-16]}
    __shared__ v2f   Bsh[KB * 32];        // 50176 B
    __shared__ float hsh[BLOCK][4];       //  4096 B

    const int tid  = threadIdx.x;
    const int wave = tid >> 5;
    const int lane = tid & 31;

    for (int e = tid; e < KB * 32; e += BLOCK) {
        int kb = e >> 5, l = e & 31;
        int n = l & 15, kh = l >> 4;
        int k = 4 * kb + 2 * kh;
        v2f b = {0.f, 0.f};
        if (n < 4) { b.x = W_red[n * KDIM + k]; b.y = W_red[n * KDIM + k + 1]; }
        Bsh[e] = b;
    }
    __syncthreads();

    // ---- fp32 WMMA GEMM: two 16-sample tiles per wave ----
    const int m  = lane & 15;     // M row within tile
    const int kh = lane >> 4;     // K half (0: k+0,1  1: k+2,3)
    const int s0 = blockIdx.x * BLOCK + wave * 32;

    const float* r0 = x + (size_t)(s0 + m)      * KDIM + 2 * kh;
    const float* r1 = x + (size_t)(s0 + 16 + m) * KDIM + 2 * kh;

    v8f c0 = {}; v8f c1 = {};
#pragma unroll 4
    for (int kb = 0; kb < KB; ++kb) {
        v2f a0 = *(const v2f*)(r0 + 4 * kb);
        v2f a1 = *(const v2f*)(r1 + 4 * kb);
        v2f b  = Bsh[kb * 32 + lane];
        c0 = __builtin_amdgcn_wmma_f32_16x16x4_f32(false, a0, false, b,
                                                   (short)0, c0, false, false);
        c1 = __builtin_amdgcn_wmma_f32_16x16x4_f32(false, a1, false, b,
                                                   (short)0, c1, false, false);
    }

    // C layout: lane L, VGPR v -> M = v + 8*(L>>4), N = L&15. Keep N<4 only.
    if (m < 4) {
        const int   mb   = kh * 8;
        const float bias = b_red[m];
#pragma unroll
        for (int v = 0; v < 8; ++v) {
            hsh[wave * 32 +      mb + v][m] = c0[v] + bias;
            hsh[wave * 32 + 16 + mb + v][m] = c1[v] + bias;
        }
    }
    __syncthreads();

    // ---- per-lane 4-qubit statevector simulation (one sample per lane) ----
    const int sample = blockIdx.x * BLOCK + tid;

    float ang[NQ];
#pragma unroll
    for (int w = 0; w < NQ; ++w)
        ang[w] = tanhf(hsh[tid][w]) * 3.14159265358979323846f;

    float sr[DIM], si[DIM];
#pragma unroll
    for (int s = 0; s < DIM; ++s) { sr[s] = 0.f; si[s] = 0.f; }
    sr[0] = 1.f;

    // RX(ang[w]) on wire w  (wire 0 = MSB -> bit position 3-w)
#pragma unroll
    for (int w = 0; w < NQ; ++w) {
        float c = cosf(0.5f * ang[w]);
        float s = sinf(0.5f * ang[w]);
        int mask = 1 << (3 - w);
#pragma unroll
        for (int i = 0; i < DIM; ++i) {
            if (!(i & mask)) {
                int j = i | mask;
                float r0_ = sr[i], i0_ = si[i], r1_ = sr[j], i1_ = si[j];
                sr[i] = c * r0_ + s * i1_;  si[i] = c * i0_ - s * r1_;
                sr[j] = c * r1_ + s * i0_;  si[j] = c * i1_ - s * r0_;
            }
        }
    }

    // NLAYERS x [RY(weights) per wire, CNOT ring]
#pragma unroll
    for (int layer = 0; layer < 2; ++layer) {
#pragma unroll
        for (int w = 0; w < NQ; ++w) {
            float th = qnn_w[w + layer * NQ];
            float c = cosf(0.5f * th);
            float s = sinf(0.5f * th);
            int mask = 1 << (3 - w);
#pragma unroll
            for (int i = 0; i < DIM; ++i) {
                if (!(i & mask)) {
                    int j = i | mask;
                    float r0_ = sr[i], i0_ = si[i], r1_ = sr[j], i1_ = si[j];
                    sr[i] = c * r0_ - s * r1_;  si[i] = c * i0_ - s * i1_;
                    sr[j] = s * r0_ + c * r1_;  si[j] = s * i0_ + c * i1_;
                }
            }
        }
#pragma unroll
        for (int w = 0; w < NQ; ++w) {
            int cm = 1 << (3 - w);
            int tm = 1 << (3 - ((w + 1) & 3));
#pragma unroll
            for (int i = 0; i < DIM; ++i) {
                if ((i & cm) && !(i & tm)) {
                    int j = i | tm;
                    float tr = sr[i]; sr[i] = sr[j]; sr[j] = tr;
                    float ti = si[i]; si[i] = si[j]; si[j] = ti;
                }
            }
        }
    }

    // <Z_w> = sum_s (+/-) |amp_s|^2
    float ev[NQ];
#pragma unroll
    for (int w = 0; w < NQ; ++w) {
        int pm = 1 << (3 - w);
        float e = 0.f;
#pragma unroll
        for (int i = 0; i < DIM; ++i) {
            float p = sr[i] * sr[i] + si[i] * si[i];
            e += (i & pm) ? -p : p;
        }
        ev[w] = e;
    }

    // output head: [4] -> [10]
    float* o = out + (size_t)sample * 10;
#pragma unroll
    for (int j = 0; j < 10; ++j) {
        float acc = b_out[j];
#pragma unroll
        for (int i = 0; i < NQ; ++i) acc += ev[i] * W_out[j * NQ + i];
        o[j] = acc;
    }
}

extern "C" void kernel_launch(void* const* d_in, const int* in_sizes, int n_in,
                              void* d_out, int out_size, void* d_ws, size_t ws_size,
                              hipStream_t stream) {
    const float* x      = (const float*)d_in[0];
    const float* W_red  = (const float*)d_in[1];
    const float* b_red  = (const float*)d_in[2];
    const float* qnn_w  = (const float*)d_in[3];
    const float* W_out  = (const float*)d_in[4];
    const float* b_out  = (const float*)d_in[5];
    float*       out    = (float*)d_out;

    const int B    = in_sizes[0] / (28 * 28);  // 131072
    const int grid = B / BLOCK;                // 512 (B divisible by 256)

    hybrid_qnn_kernel<<<grid, BLOCK, 0, stream>>>(x, W_red, b_red, qnn_w,
                                                  W_out, b_out, out);
}